// Attention_51994874085842
// MI455X (gfx1250) — compile-verified
//
#include <hip/hip_runtime.h>
#include <math.h>

typedef __attribute__((ext_vector_type(2))) float v2f;
typedef __attribute__((ext_vector_type(8))) float v8f;

#define S_LEN 1024
#define D_HEAD 64
#define N_BH 64
#define N_TOT (1u << 20)     /* S*S elements per (b,h) */
#define N_BINS (1 << 20)     /* top-20-bit key histogram */
#define BIN_SHIFT 12
#define EPS_NRM 1e-5f

// Monotone map: float -> u32 such that integer order == float order (ascending).
__device__ __forceinline__ unsigned key_of(float f) {
  unsigned u = __float_as_uint(f);
  return (u & 0x80000000u) ? ~u : (u | 0x80000000u);
}

// ---- row norms of Q and K: qn/kn[bh*1024 + row] = sqrt(sum d) + eps --------
__global__ void norms_kernel(const float* __restrict__ Q, const float* __restrict__ K,
                             float* __restrict__ qn, float* __restrict__ kn) {
  int idx = blockIdx.x * blockDim.x + threadIdx.x;   // 0..131071
  int row = idx & 65535;
  const float* p = ((idx < 65536) ? Q : K) + (size_t)row * D_HEAD;
  float s = 0.f;
  for (int d = 0; d < D_HEAD; ++d) { float v = p[d]; s += v * v; }
  float r = sqrtf(s) + EPS_NRM;
  if (idx < 65536) qn[row] = r; else kn[row] = r;
}

// ---- cosine score via f32 WMMA; emit monotone keys + histogram -------------
__global__ void score_key_hist_kernel(const float* __restrict__ Q, const float* __restrict__ K,
                                      const float* __restrict__ qn, const float* __restrict__ kn,
                                      unsigned* __restrict__ skey, unsigned* __restrict__ hist,
                                      int bh) {
  const int wave = threadIdx.x >> 5;
  const int lane = threadIdx.x & 31;
  const int tile = blockIdx.x * 4 + wave;            // 0..4095 (64x64 tiles of 16x16)
  const int m0 = (tile >> 6) << 4;
  const int n0 = (tile & 63) << 4;
  const int l = lane & 15;
  const int h = lane >> 4;                           // lane half selects K pair
  const float* Qb = Q + (size_t)bh * S_LEN * D_HEAD;
  const float* Kb = K + (size_t)bh * S_LEN * D_HEAD;

  v8f acc = {};
  for (int k0 = 0; k0 < D_HEAD; k0 += 4) {
    v2f a, b;
    a.x = Qb[(m0 + l) * D_HEAD + k0 + 2 * h];
    a.y = Qb[(m0 + l) * D_HEAD + k0 + 2 * h + 1];
    b.x = Kb[(n0 + l) * D_HEAD + k0 + 2 * h];
    b.y = Kb[(n0 + l) * D_HEAD + k0 + 2 * h + 1];
    acc = __builtin_amdgcn_wmma_f32_16x16x4_f32(false, a, false, b, (short)0, acc, false, false);
  }

  const float* qnb = qn + (size_t)bh * S_LEN;
  const float* knb = kn + (size_t)bh * S_LEN;
  float kln = knb[n0 + l];
  for (int r = 0; r < 8; ++r) {
    int row = m0 + r + (h << 3);
    int col = n0 + l;
    float val = acc[r] / (qnb[row] * kln);
    unsigned kk = key_of(val);
    skey[(size_t)row * S_LEN + col] = kk;
    atomicAdd(&hist[kk >> BIN_SHIFT], 1u);
  }
}

__global__ void zero_u32(unsigned* __restrict__ p, int n) {
  int i = blockIdx.x * blockDim.x + threadIdx.x;
  if (i < n) p[i] = 0u;
}

// ---- suffix-sum of histogram: ES[b] = #elements in bins > b ----------------
__global__ void hist_blocksum_kernel(const unsigned* __restrict__ hist, unsigned* __restrict__ bsum) {
  __shared__ unsigned sh[256];
  int t = threadIdx.x;
  sh[t] = hist[(size_t)blockIdx.x * 256 + t];
  __syncthreads();
  for (int o = 128; o > 0; o >>= 1) { if (t < o) sh[t] += sh[t + o]; __syncthreads(); }
  if (t == 0) bsum[blockIdx.x] = sh[0];
}

__global__ void suffix_top_kernel(const unsigned* __restrict__ bsum, unsigned* __restrict__ bsuf) {
  unsigned run = 0;
  for (int i = 4095; i >= 0; --i) { bsuf[i] = run; run += bsum[i]; }
}

__global__ void bin_offsets_kernel(const unsigned* __restrict__ hist, const unsigned* __restrict__ bsuf,
                                   unsigned* __restrict__ ES, unsigned* __restrict__ cursor) {
  __shared__ unsigned sc[256];
  __shared__ unsigned sf[256];
  int t = threadIdx.x;
  int b = blockIdx.x * 256 + t;
  sc[t] = hist[b];
  __syncthreads();
  if (t == 0) { unsigned run = 0; for (int i = 255; i >= 0; --i) { sf[i] = run; run += sc[i]; } }
  __syncthreads();
  unsigned es = bsuf[blockIdx.x] + sf[t];
  ES[b] = es;
  cursor[b] = N_TOT - es - sc[t];                    // bin start (ascending layout)
}

// ---- scatter (key, original index) into per-bin contiguous regions ---------
__global__ void scatter_kernel(const unsigned* __restrict__ skey, unsigned* __restrict__ cursor,
                               unsigned* __restrict__ gkey, unsigned* __restrict__ gidx) {
  unsigned e = blockIdx.x * blockDim.x + threadIdx.x;
  unsigned k = skey[e];
  unsigned pos = atomicAdd(&cursor[k >> BIN_SHIFT], 1u);
  gkey[pos] = k;
  gidx[pos] = e;
}

// ---- exact descending rank = ES[bin] + (#greater-in-bin, stable ties) ------
__global__ void rank_weight_kernel(const unsigned* __restrict__ gkey, const unsigned* __restrict__ gidx,
                                   const unsigned* __restrict__ ES, const unsigned* __restrict__ hist,
                                   float* __restrict__ wgt) {
  unsigned s = blockIdx.x * blockDim.x + threadIdx.x;
  unsigned k = gkey[s];
  unsigned e = gidx[s];
  unsigned b = k >> BIN_SHIFT;
  unsigned base = ES[b];
  unsigned c = hist[b];
  unsigned st = N_TOT - base - c;
  unsigned cnt = 0;
  for (unsigned j = st; j < st + c; ++j) {
    unsigned kj = gkey[j];
    if (kj > k) cnt++;
    else if (kj == k && gidx[j] < e) cnt++;          // stable tie-break (argsort order)
  }
  unsigned rank = base + cnt;
  wgt[e] = -logf((float)(rank + 1) * (1.0f / (float)N_TOT));
}

__global__ void rowsum_kernel(const float* __restrict__ wgt, float* __restrict__ rowsum) {
  __shared__ float sh[256];
  int q = blockIdx.x;
  int t = threadIdx.x;
  const float* r = wgt + (size_t)q * S_LEN;
  float s = 0.f;
  for (int k = t; k < S_LEN; k += 256) s += r[k];
  sh[t] = s; __syncthreads();
  for (int o = 128; o > 0; o >>= 1) { if (t < o) sh[t] += sh[t + o]; __syncthreads(); }
  if (t == 0) rowsum[q] = sh[0];
}

// ---- out = (W @ V) / rowsum via f32 WMMA -----------------------------------
__global__ void out_wmma_kernel(const float* __restrict__ wgt, const float* __restrict__ V,
                                const float* __restrict__ rowsum, float* __restrict__ out, int bh) {
  const int wave = threadIdx.x >> 5;
  const int lane = threadIdx.x & 31;
  const int tile = blockIdx.x * 4 + wave;            // 0..255 (64x4 tiles of 16x16)
  const int m0 = (tile >> 2) << 4;
  const int n0 = (tile & 3) << 4;
  const int l = lane & 15;
  const int h = lane >> 4;
  const float* Vb = V + (size_t)bh * S_LEN * D_HEAD;

  v8f acc = {};
  for (int k0 = 0; k0 < S_LEN; k0 += 4) {
    v2f a, b;
    a.x = wgt[(size_t)(m0 + l) * S_LEN + k0 + 2 * h];
    a.y = wgt[(size_t)(m0 + l) * S_LEN + k0 + 2 * h + 1];
    b.x = Vb[(k0 + 2 * h) * D_HEAD + n0 + l];
    b.y = Vb[(k0 + 2 * h + 1) * D_HEAD + n0 + l];
    acc = __builtin_amdgcn_wmma_f32_16x16x4_f32(false, a, false, b, (short)0, acc, false, false);
  }

  float* ob = out + (size_t)bh * S_LEN * D_HEAD;
  for (int r = 0; r < 8; ++r) {
    int row = m0 + r + (h << 3);
    int col = n0 + l;
    ob[(size_t)row * D_HEAD + col] = acc[r] / rowsum[row];
  }
}

extern "C" void kernel_launch(void* const* d_in, const int* in_sizes, int n_in,
                              void* d_out, int out_size, void* d_ws, size_t ws_size,
                              hipStream_t stream) {
  const float* Q = (const float*)d_in[0];
  const float* K = (const float*)d_in[1];
  const float* V = (const float*)d_in[2];
  float* out = (float*)d_out;
  char* ws = (char*)d_ws;

  size_t off = 0;
  float*    qn     = (float*)(ws + off);    off += (size_t)65536 * 4;
  float*    kn     = (float*)(ws + off);    off += (size_t)65536 * 4;
  unsigned* skey   = (unsigned*)(ws + off); off += (size_t)N_TOT * 4;
  unsigned* hist   = (unsigned*)(ws + off); off += (size_t)N_BINS * 4;
  unsigned* ES     = (unsigned*)(ws + off); off += (size_t)N_BINS * 4;
  unsigned* cursor = (unsigned*)(ws + off); off += (size_t)N_BINS * 4;
  unsigned* gkey   = (unsigned*)(ws + off); off += (size_t)N_TOT * 4;
  unsigned* gidx   = (unsigned*)(ws + off); off += (size_t)N_TOT * 4;
  float*    wgt    = (float*)(ws + off);    off += (size_t)N_TOT * 4;
  float*    rowsum = (float*)(ws + off);    off += (size_t)S_LEN * 4;
  unsigned* bsum   = (unsigned*)(ws + off); off += (size_t)4096 * 4;
  unsigned* bsuf   = (unsigned*)(ws + off); off += (size_t)4096 * 4;

  norms_kernel<<<512, 256, 0, stream>>>(Q, K, qn, kn);

  for (int bh = 0; bh < N_BH; ++bh) {
    zero_u32<<<N_BINS / 256, 256, 0, stream>>>(hist, N_BINS);
    score_key_hist_kernel<<<1024, 128, 0, stream>>>(Q, K, qn, kn, skey, hist, bh);
    hist_blocksum_kernel<<<4096, 256, 0, stream>>>(hist, bsum);
    suffix_top_kernel<<<1, 1, 0, stream>>>(bsum, bsuf);
    bin_offsets_kernel<<<4096, 256, 0, stream>>>(hist, bsuf, ES, cursor);
    scatter_kernel<<<4096, 256, 0, stream>>>(skey, cursor, gkey, gidx);
    rank_weight_kernel<<<4096, 256, 0, stream>>>(gkey, gidx, ES, hist, wgt);
    rowsum_kernel<<<S_LEN, 256, 0, stream>>>(wgt, rowsum);
    out_wmma_kernel<<<64, 128, 0, stream>>>(wgt, V, rowsum, out, bh);
  }

  (void)in_sizes; (void)n_in; (void)out_size; (void)ws_size;
}